// SqueezeExcitation_85212151153392
// MI455X (gfx1250) — compile-verified
//
#include <hip/hip_runtime.h>
#include <hip/hip_bf16.h>
#include <math.h>

// Problem constants (fixed by the reference setup)
#define B   32
#define C   256
#define R   16          // C / 16
#define HW  16384       // 128*128
#define BC  (B * C)     // 8192

typedef __attribute__((ext_vector_type(2))) float v2f;
typedef __attribute__((ext_vector_type(8))) float v8f;

// ---------------------------------------------------------------------------
// Kernel 1: global average pool.  One 256-thread block per (b,c) plane.
// ---------------------------------------------------------------------------
__global__ __launch_bounds__(256) void se_pool(const float* __restrict__ x,
                                               float* __restrict__ pooled) {
    __shared__ float red[256];
    const int bc = blockIdx.x;                       // 0..BC-1
    const float4* xp = (const float4*)(x + (size_t)bc * HW);
    float s = 0.0f;
#pragma unroll 4
    for (int i = threadIdx.x; i < HW / 4; i += 256) {
        float4 v = xp[i];
        s += v.x + v.y + v.z + v.w;
    }
    red[threadIdx.x] = s;
    __syncthreads();
#pragma unroll
    for (int off = 128; off > 0; off >>= 1) {
        if (threadIdx.x < off) red[threadIdx.x] += red[threadIdx.x + off];
        __syncthreads();
    }
    if (threadIdx.x == 0) pooled[bc] = red[0] * (1.0f / (float)HW);
}

// ---------------------------------------------------------------------------
// Kernel 2: SE MLP, single wave32, all matmuls via V_WMMA_F32_16X16X4_F32.
//
// Fragment layouts (ISA 7.12.2, wave32):
//   A (16x4 f32, 2 VGPRs):  lane L, vgpr v -> A[L%16][2*(L/16) + v]
//   B (4x16 f32, 2 VGPRs):  lane L, vgpr v -> B[2*(L/16) + v][L%16]
//   C/D (16x16 f32, 8 VGPRs): lane L, vgpr v -> D[v + 8*(L/16)][L%16]
// Each lane's two A/B values are K-contiguous -> float2 global/LDS loads.
// ---------------------------------------------------------------------------
__global__ __launch_bounds__(32) void se_mlp(const float* __restrict__ pooled, // [B, C]
                                             const float* __restrict__ w1,     // [R, C]
                                             const float* __restrict__ w2,     // [C, R]
                                             float* __restrict__ gate) {       // [B, C]
    __shared__ __align__(16) float h_lds[B * R];     // h activation, row-major [32][16]

    const int lane = threadIdx.x;
    const int half = lane >> 4;                      // 0: lanes 0-15, 1: lanes 16-31
    const int l16  = lane & 15;

    // ---------------- h = relu(pooled @ w1^T) : M=32, N=16, K=256 ----------
    v8f acc0 = {};                                   // rows 0..15
    v8f acc1 = {};                                   // rows 16..31
    for (int k0 = 0; k0 < C; k0 += 4) {
        const int kk = k0 + 2 * half;
        v2f a0 = *(const v2f*)(pooled + (size_t)l16        * C + kk);
        v2f a1 = *(const v2f*)(pooled + (size_t)(l16 + 16) * C + kk);
        v2f bf = *(const v2f*)(w1     + (size_t)l16        * C + kk); // w1^T[k][n] = w1[n][k]
        acc0 = __builtin_amdgcn_wmma_f32_16x16x4_f32(false, a0, false, bf,
                                                     (short)0, acc0, false, false);
        acc1 = __builtin_amdgcn_wmma_f32_16x16x4_f32(false, a1, false, bf,
                                                     (short)0, acc1, false, false);
    }
    // ReLU, then stage h into LDS row-major so it can be re-read in A-layout.
#pragma unroll
    for (int v = 0; v < 8; ++v) {
        const int m = v + 8 * half;
        h_lds[m * R + l16]        = fmaxf(acc0[v], 0.0f);
        h_lds[(m + 16) * R + l16] = fmaxf(acc1[v], 0.0f);
    }
    __syncthreads();

    // ---------------- gate = sigmoid(h @ w2^T) : M=32, N=256, K=16 ---------
    for (int nt = 0; nt < C / 16; ++nt) {
        const int n0 = nt * 16;
        v8f g0 = {};
        v8f g1 = {};
#pragma unroll
        for (int ks = 0; ks < 4; ++ks) {
            const int kk = 4 * ks + 2 * half;
            v2f a0 = *(const v2f*)(h_lds + l16        * R + kk);
            v2f a1 = *(const v2f*)(h_lds + (l16 + 16) * R + kk);
            v2f bf = *(const v2f*)(w2 + (size_t)(n0 + l16) * R + kk); // w2^T[k][n] = w2[n][k]
            g0 = __builtin_amdgcn_wmma_f32_16x16x4_f32(false, a0, false, bf,
                                                       (short)0, g0, false, false);
            g1 = __builtin_amdgcn_wmma_f32_16x16x4_f32(false, a1, false, bf,
                                                       (short)0, g1, false, false);
        }
#pragma unroll
        for (int v = 0; v < 8; ++v) {
            const int m = v + 8 * half;
            gate[m * C + n0 + l16]        = 1.0f / (1.0f + __expf(-g0[v]));
            gate[(m + 16) * C + n0 + l16] = 1.0f / (1.0f + __expf(-g1[v]));
        }
    }
}

// ---------------------------------------------------------------------------
// Kernel 3: out = x * gate[b,c], streaming float4.  gate (32 KB) lives in L2.
// ---------------------------------------------------------------------------
__global__ __launch_bounds__(256) void se_scale(const float* __restrict__ x,
                                                const float* __restrict__ gate,
                                                float* __restrict__ out,
                                                int n4) {
    const int idx4 = blockIdx.x * 256 + threadIdx.x;     // float4 index
    if (idx4 >= n4) return;
    const float g = gate[idx4 >> 12];                    // (idx4*4) / HW, HW/4 = 4096
    float4 v = ((const float4*)x)[idx4];
    v.x *= g; v.y *= g; v.z *= g; v.w *= g;
    ((float4*)out)[idx4] = v;
}

// ---------------------------------------------------------------------------
extern "C" void kernel_launch(void* const* d_in, const int* in_sizes, int n_in,
                              void* d_out, int out_size, void* d_ws, size_t ws_size,
                              hipStream_t stream) {
    const float* x  = (const float*)d_in[0];   // [B, C, H, W]
    const float* w1 = (const float*)d_in[1];   // [R, C]
    const float* w2 = (const float*)d_in[2];   // [C, R]
    float* out = (float*)d_out;

    float* pooled = (float*)d_ws;              // BC floats
    float* gate   = pooled + BC;               // BC floats

    se_pool<<<BC, 256, 0, stream>>>(x, pooled);
    se_mlp<<<1, 32, 0, stream>>>(pooled, w1, w2, gate);

    const int n4 = (B * C * HW) / 4;           // 33,554,432 float4s
    se_scale<<<n4 / 256, 256, 0, stream>>>(x, gate, out, n4);
}